// HMM_26018911879523
// MI455X (gfx1250) — compile-verified
//
#include <hip/hip_runtime.h>
#include <math.h>

// ---------------------------------------------------------------------------
// Kalman-filter HMM forward pass for MI455X (gfx1250), single persistent WGP.
// All state LDS-resident; fp32 WMMA (v_wmma_f32_16x16x4_f32) for all GEMMs.
// ---------------------------------------------------------------------------

#define T_STEPS 2048
#define OD      128            // observation dim (2 * 64 sensors)
#define DIM     132            // augmented latent dim (OD + 4)
#define DP      144            // padded dim (multiple of 16) for P / W strides
#define SSTR    132            // row stride of S / L (128x128 matrix)
#define NT      256            // threads per block (8 wave32)
#define NW      8

#define LOG2PI_F 1.8378770664093453f

typedef __attribute__((ext_vector_type(2))) float v2f;
typedef __attribute__((ext_vector_type(8))) float v8f;

// LDS layout (float offsets)
#define OFF_P   0                       // P   : DP x DP     (20736)
#define OFF_W   (OFF_P + DP*DP)         // W   : DP x DP     (20736)
#define OFF_L   (OFF_W + DP*DP)         // S/L : OD x SSTR   (16896)
#define OFF_M   (OFF_L + OD*SSTR)       // m   : DP
#define OFF_V   (OFF_M + DP)            // w/v : OD
#define OFF_H   (OFF_V + OD)            // hob : 4 x OD
#define OFF_S   (OFF_H + 4*OD)          // scalars: [0] = log-likelihood
#define SMEM_FLOATS (OFF_S + 16)

// ---------------------------------------------------------------------------
// 16x16 fp32 tile:  C -= A(16xK) * B(Kx16), all operands in LDS.
// A(m,k) = A[m*aRS + k*aCS], B(k,n) = B[k*bRS + n*bCS] (strides = free transpose)
// Product accumulated into a zero chain; single subtract at store (avoids
// per-fragment sign flips). Must be entered wave-uniformly (EXEC all ones).
// Fragment layouts per CDNA5 ISA 7.12.2 (f32 16x16x4).
// ---------------------------------------------------------------------------
__device__ __forceinline__ void wmma_tile_sub(float* __restrict__ C, int cS,
                                              const float* __restrict__ A, int aRS, int aCS,
                                              const float* __restrict__ B, int bRS, int bCS,
                                              int Klen, int lane)
{
    const int half = lane >> 4;
    const int l16  = lane & 15;

    v8f acc;
#pragma unroll
    for (int u = 0; u < 8; ++u) acc[u] = 0.0f;

    for (int k0 = 0; k0 < Klen; k0 += 4) {
        const int ka = k0 + 2 * half;
        v2f av, bv;
        av[0] = A[l16 * aRS + (ka + 0) * aCS];
        av[1] = A[l16 * aRS + (ka + 1) * aCS];
        bv[0] = B[(ka + 0) * bRS + l16 * bCS];
        bv[1] = B[(ka + 1) * bRS + l16 * bCS];
        acc = __builtin_amdgcn_wmma_f32_16x16x4_f32(false, av, false, bv,
                                                    (short)0, acc, false, false);
    }

#pragma unroll
    for (int u = 0; u < 8; ++u) {
        const int r = (u + 8 * half) * cS + l16;
        C[r] = C[r] - acc[u];
    }
}

// Dual-tile variant: C0 -= A*B0, C1 -= A*B1 with the A fragment loaded once.
// Two independent accumulators -> back-to-back WMMAs with no RAW hazard.
__device__ __forceinline__ void wmma_tile_sub_dual(float* __restrict__ C0,
                                                   float* __restrict__ C1, int cS,
                                                   const float* __restrict__ A, int aRS, int aCS,
                                                   const float* __restrict__ B0,
                                                   const float* __restrict__ B1, int bRS, int bCS,
                                                   int Klen, int lane)
{
    const int half = lane >> 4;
    const int l16  = lane & 15;

    v8f acc0, acc1;
#pragma unroll
    for (int u = 0; u < 8; ++u) { acc0[u] = 0.0f; acc1[u] = 0.0f; }

    for (int k0 = 0; k0 < Klen; k0 += 4) {
        const int ka = k0 + 2 * half;
        v2f av, bv0, bv1;
        av[0]  = A[l16 * aRS + (ka + 0) * aCS];
        av[1]  = A[l16 * aRS + (ka + 1) * aCS];
        bv0[0] = B0[(ka + 0) * bRS + l16 * bCS];
        bv0[1] = B0[(ka + 1) * bRS + l16 * bCS];
        bv1[0] = B1[(ka + 0) * bRS + l16 * bCS];
        bv1[1] = B1[(ka + 1) * bRS + l16 * bCS];
        acc0 = __builtin_amdgcn_wmma_f32_16x16x4_f32(false, av, false, bv0,
                                                     (short)0, acc0, false, false);
        acc1 = __builtin_amdgcn_wmma_f32_16x16x4_f32(false, av, false, bv1,
                                                     (short)0, acc1, false, false);
    }

#pragma unroll
    for (int u = 0; u < 8; ++u) {
        const int r = (u + 8 * half) * cS + l16;
        C0[r] = C0[r] - acc0[u];
        C1[r] = C1[r] - acc1[u];
    }
}

extern "C" __global__ void __launch_bounds__(NT)
hmm_kalman_gfx1250(const float* __restrict__ track,        // [T_STEPS*2]
                   const float* __restrict__ bias_scales,  // [2]
                   const float* __restrict__ obs_noise_p,  // [1]
                   const float* __restrict__ trans_noise_p,// [1]
                   float* __restrict__ out)                // [1 + DIM*DIM]
{
    extern __shared__ float s[];
    float* P    = s + OFF_P;
    float* W    = s + OFF_W;
    float* L    = s + OFF_L;
    float* m    = s + OFF_M;
    float* wv   = s + OFF_V;
    float* hob  = s + OFF_H;
    float* scal = s + OFF_S;

    const int tid  = threadIdx.x;
    const int lane = tid & 31;
    // Force the wave id into an SGPR so tile-ownership tests compile to scalar
    // s_cmp/s_cbranch (EXEC stays all-ones for WMMA, as the ISA requires).
    const int wave = __builtin_amdgcn_readfirstlane(tid >> 5);

    const float on2 = obs_noise_p[0] * obs_noise_p[0];
    const float tn2 = trans_noise_p[0] * trans_noise_p[0];

    // ---------------- init: P0 = diag(bias_var, 1,1,1,1), m0 = 0 ------------
    for (int i = tid; i < DP * DP; i += NT) { P[i] = 0.0f; W[i] = 0.0f; }
    for (int i = tid; i < DP; i += NT) m[i] = 0.0f;
    __syncthreads();
    for (int i = tid; i < OD; i += NT)
        P[i * DP + i] = bias_scales[i & 1];        // interleaved (S,2) bias var
    if (tid < 4)
        P[(OD + tid) * DP + (OD + tid)] = 1.0f;
    // Hobs rows, reproducing eye(4,2).expand(S,4,2).transpose(2,1,0).reshape(4,OD)
    for (int idx = tid; idx < 4 * OD; idx += NT) {
        int r = idx >> 7, c = idx & 127;
        int f   = r * OD + c;       // flat over (2,4,64)
        int j   = f >> 8;           // / 256
        int rem = f & 255;
        int i2  = rem >> 6;         // / 64
        hob[idx] = (i2 == j) ? 1.0f : 0.0f;
    }
    if (tid == 0) scal[0] = 0.0f;
    __syncthreads();

    for (int t = 0; t < T_STEPS; ++t) {
        // ========== predict: Pp = A P A^T + Q, mp = A m (in place) ==========
        for (int i = tid; i < OD; i += NT) {
            P[i * DP + OD + 0] += P[i * DP + OD + 2];     // cross block * F^T
            P[i * DP + OD + 1] += P[i * DP + OD + 3];
            P[(OD + 0) * DP + i] += P[(OD + 2) * DP + i]; // F * cross block
            P[(OD + 1) * DP + i] += P[(OD + 3) * DP + i];
        }
        if (tid == 0) {
            const float Qm[4][4] = {{1.0f/3.0f, 0.0f, 0.5f, 0.0f},
                                    {0.0f, 1.0f/3.0f, 0.0f, 0.5f},
                                    {0.5f, 0.0f, 1.0f, 0.0f},
                                    {0.0f, 0.5f, 0.0f, 1.0f}};
            float p[4][4], fp[4][4];
            for (int a = 0; a < 4; ++a)
                for (int b = 0; b < 4; ++b)
                    p[a][b] = P[(OD + a) * DP + (OD + b)];
            for (int b = 0; b < 4; ++b) {                 // F * p
                fp[0][b] = p[0][b] + p[2][b];
                fp[1][b] = p[1][b] + p[3][b];
                fp[2][b] = p[2][b];
                fp[3][b] = p[3][b];
            }
            for (int a = 0; a < 4; ++a) {                 // (F p) F^T + Q
                float r0 = fp[a][0] + fp[a][2];
                float r1 = fp[a][1] + fp[a][3];
                P[(OD + a) * DP + OD + 0] = r0       + tn2 * Qm[a][0];
                P[(OD + a) * DP + OD + 1] = r1       + tn2 * Qm[a][1];
                P[(OD + a) * DP + OD + 2] = fp[a][2] + tn2 * Qm[a][2];
                P[(OD + a) * DP + OD + 3] = fp[a][3] + tn2 * Qm[a][3];
            }
            m[OD + 0] += m[OD + 2];                       // mp = A m
            m[OD + 1] += m[OD + 3];
            if (t + 1 < T_STEPS)
                __builtin_prefetch(track + 2 * (t + 1), 0, 1);
        }
        __syncthreads();

        // ========== W = Haug * Pp (128 x 132, pad cols zeroed) ==============
        for (int idx = tid; idx < OD * DP; idx += NT) {
            int i = idx / DP, j = idx - (idx / DP) * DP;
            float v = 0.0f;
            if (j < DIM) {
                v = P[i * DP + j]
                  + hob[0 * OD + i] * P[(OD + 0) * DP + j]
                  + hob[1 * OD + i] * P[(OD + 1) * DP + j]
                  + hob[2 * OD + i] * P[(OD + 2) * DP + j]
                  + hob[3 * OD + i] * P[(OD + 3) * DP + j];
            }
            W[i * DP + j] = v;
        }
        // innovation v = y - Haug * mp  (y interleaved per sensor)
        for (int i = tid; i < OD; i += NT) {
            float y = track[2 * t + (i & 1)];
            wv[i] = y - (m[i]
                         + hob[0 * OD + i] * m[OD + 0]
                         + hob[1 * OD + i] * m[OD + 1]
                         + hob[2 * OD + i] * m[OD + 2]
                         + hob[3 * OD + i] * m[OD + 3]);
        }
        __syncthreads();

        // ========== S = W * Haug^T + R (128 x 128) ==========================
        for (int idx = tid; idx < OD * OD; idx += NT) {
            int i = idx >> 7, j = idx & 127;
            float v = W[i * DP + j]
                    + W[i * DP + OD + 0] * hob[0 * OD + j]
                    + W[i * DP + OD + 1] * hob[1 * OD + j]
                    + W[i * DP + OD + 2] * hob[2 * OD + j]
                    + W[i * DP + OD + 3] * hob[3 * OD + j];
            if (i == j) v += on2;
            L[i * SSTR + j] = v;
        }
        __syncthreads();

        // ========== blocked Cholesky S = L L^T (panels of 16) ===============
        for (int kb = 0; kb < 8; ++kb) {
            const int c0 = kb * 16;
            for (int kk = 0; kk < 16; ++kk) {
                const int k = c0 + kk;
                if (tid == 0) L[k * SSTR + k] = sqrtf(L[k * SSTR + k]);
                __syncthreads();
                const float dk = L[k * SSTR + k];
                for (int i = k + 1 + tid; i < OD; i += NT)
                    L[i * SSTR + k] /= dk;
                __syncthreads();
                for (int idx = tid; idx < 16 * OD; idx += NT) {
                    int jj = idx >> 7, i = idx & 127;
                    int j = c0 + jj;
                    if (j > k && i >= j)
                        L[i * SSTR + j] -= L[i * SSTR + k] * L[j * SSTR + k];
                }
                __syncthreads();
            }
            // trailing SYRK: L[tr,tr] -= Lp * Lp^T  (WMMA, lower tiles)
            int cnt = 0;
            for (int rb = kb + 1; rb < 8; ++rb)
                for (int cb = kb + 1; cb <= rb; ++cb, ++cnt)
                    if ((cnt & (NW - 1)) == wave)
                        wmma_tile_sub(&L[(rb * 16) * SSTR + cb * 16], SSTR,
                                      &L[(rb * 16) * SSTR + c0], SSTR, 1,
                                      &L[(cb * 16) * SSTR + c0], 1, SSTR, // B = Lp^T
                                      16, lane);
            __syncthreads();
        }

        // ========== forward TRSM: W = L^{-1} W  (blocked 16) ================
        for (int rb = 0; rb < 8; ++rb) {
            const int r0 = rb * 16;
            for (int rr = 0; rr < 16; ++rr) {
                const int i = r0 + rr;
                const float dii = L[i * SSTR + i];
                for (int j = tid; j < DP; j += NT) {
                    float acc = W[i * DP + j];
                    for (int k = r0; k < i; ++k)
                        acc -= L[i * SSTR + k] * W[k * DP + j];
                    W[i * DP + j] = acc / dii;
                }
                __syncthreads();
            }
            // trailing: W[rb2,:] -= L[rb2,rb] * W[rb,:]  (dual-tile WMMA,
            // A = 16x16 L block reused across the 9 column tiles)
            int cnt = 0;
            for (int rb2 = rb + 1; rb2 < 8; ++rb2) {
                const float* Ablk = &L[(rb2 * 16) * SSTR + r0];
                for (int ctp = 0; ctp < 5; ++ctp, ++cnt) {
                    if ((cnt & (NW - 1)) != wave) continue;
                    const int ct0 = 2 * ctp;
                    if (ct0 + 1 < 9)
                        wmma_tile_sub_dual(&W[(rb2 * 16) * DP + ct0 * 16],
                                           &W[(rb2 * 16) * DP + (ct0 + 1) * 16], DP,
                                           Ablk, SSTR, 1,
                                           &W[r0 * DP + ct0 * 16],
                                           &W[r0 * DP + (ct0 + 1) * 16], DP, 1,
                                           16, lane);
                    else
                        wmma_tile_sub(&W[(rb2 * 16) * DP + ct0 * 16], DP,
                                      Ablk, SSTR, 1,
                                      &W[r0 * DP + ct0 * 16], DP, 1,
                                      16, lane);
                }
            }
            __syncthreads();
        }

        // ========== w = L^{-1} v (column sweep) =============================
        for (int k = 0; k < OD; ++k) {
            if (tid == 0) wv[k] /= L[k * SSTR + k];
            __syncthreads();
            const float wk = wv[k];
            for (int i = k + 1 + tid; i < OD; i += NT)
                wv[i] -= L[i * SSTR + k] * wk;
            __syncthreads();
        }

        // ========== log-likelihood + mean update ============================
        if (tid == 0) {
            float logdet = 0.0f, quad = 0.0f;
            for (int i = 0; i < OD; ++i) {
                logdet += logf(L[i * SSTR + i]);
                quad   += wv[i] * wv[i];
            }
            scal[0] -= 0.5f * ((float)OD * LOG2PI_F + 2.0f * logdet + quad);
        }
        for (int j = tid; j < DIM; j += NT) {
            float acc = m[j];
            for (int i = 0; i < OD; ++i)
                acc += W[i * DP + j] * wv[i];
            m[j] = acc;
        }
        __syncthreads();

        // ========== P = Pp - W^T W  (dual-tile WMMA, K = 128) ===============
        {
            int cnt = 0;
            for (int ti = 0; ti < 9; ++ti) {
                const float* At = &W[ti * 16];            // A = W^T (aRS=1, aCS=DP)
                for (int tjp = 0; tjp < 5; ++tjp, ++cnt) {
                    if ((cnt & (NW - 1)) != wave) continue;
                    const int tj0 = 2 * tjp;
                    if (tj0 + 1 < 9)
                        wmma_tile_sub_dual(&P[(ti * 16) * DP + tj0 * 16],
                                           &P[(ti * 16) * DP + (tj0 + 1) * 16], DP,
                                           At, 1, DP,
                                           &W[tj0 * 16],
                                           &W[(tj0 + 1) * 16], DP, 1,
                                           OD, lane);
                    else
                        wmma_tile_sub(&P[(ti * 16) * DP + tj0 * 16], DP,
                                      At, 1, DP,
                                      &W[tj0 * 16], DP, 1,
                                      OD, lane);
                }
            }
        }
        __syncthreads();

        // ========== symmetrize ==============================================
        for (int idx = tid; idx < DP * DP; idx += NT) {
            int i = idx / DP, j = idx - (idx / DP) * DP;
            if (j > i) {
                float a = 0.5f * (P[i * DP + j] + P[j * DP + i]);
                P[i * DP + j] = a;
                P[j * DP + i] = a;
            }
        }
        __syncthreads();
    }

    // ================= prec = inv(P_T) via Cholesky =========================
    for (int k = 0; k < DIM; ++k) {
        if (tid == 0) P[k * DP + k] = sqrtf(P[k * DP + k]);
        __syncthreads();
        const float dk = P[k * DP + k];
        for (int i = k + 1 + tid; i < DIM; i += NT)
            P[i * DP + k] /= dk;
        __syncthreads();
        for (int idx = tid; idx < DIM * DIM; idx += NT) {
            int i = idx / DIM, j = idx - (idx / DIM) * DIM;
            if (j > k && i >= j)
                P[i * DP + j] -= P[i * DP + k] * P[j * DP + k];
        }
        __syncthreads();
    }
    // W = I
    for (int idx = tid; idx < DIM * DIM; idx += NT) {
        int i = idx / DIM, j = idx - (idx / DIM) * DIM;
        W[i * DP + j] = (i == j) ? 1.0f : 0.0f;
    }
    __syncthreads();
    // forward: W = L^{-1}
    for (int k = 0; k < DIM; ++k) {
        const float dk = P[k * DP + k];
        for (int j = tid; j < DIM; j += NT) W[k * DP + j] /= dk;
        __syncthreads();
        for (int idx = tid; idx < DIM * DIM; idx += NT) {
            int i = idx / DIM, j = idx - (idx / DIM) * DIM;
            if (i > k) W[i * DP + j] -= P[i * DP + k] * W[k * DP + j];
        }
        __syncthreads();
    }
    // backward: W = L^{-T} W  => W = P^{-1}
    for (int k = DIM - 1; k >= 0; --k) {
        const float dk = P[k * DP + k];
        for (int j = tid; j < DIM; j += NT) W[k * DP + j] /= dk;
        __syncthreads();
        for (int idx = tid; idx < DIM * DIM; idx += NT) {
            int i = idx / DIM, j = idx - (idx / DIM) * DIM;
            if (i < k) W[i * DP + j] -= P[k * DP + i] * W[k * DP + j];
        }
        __syncthreads();
    }

    // ================= write outputs: [ll, prec(132x132)] ===================
    if (tid == 0) out[0] = scal[0];
    for (int idx = tid; idx < DIM * DIM; idx += NT) {
        int i = idx / DIM, j = idx - (idx / DIM) * DIM;
        out[1 + idx] = W[i * DP + j];
    }
}

extern "C" void kernel_launch(void* const* d_in, const int* in_sizes, int n_in,
                              void* d_out, int out_size, void* d_ws, size_t ws_size,
                              hipStream_t stream)
{
    (void)in_sizes; (void)n_in; (void)out_size; (void)d_ws; (void)ws_size;
    const float* track       = (const float*)d_in[0];
    const float* bias_scales = (const float*)d_in[1];
    const float* obs_noise   = (const float*)d_in[2];
    const float* trans_noise = (const float*)d_in[3];
    float* out = (float*)d_out;

    const size_t smem = (size_t)SMEM_FLOATS * sizeof(float);   // ~231 KB (< 320 KB/WGP)
    (void)hipFuncSetAttribute(reinterpret_cast<const void*>(hmm_kalman_gfx1250),
                              hipFuncAttributeMaxDynamicSharedMemorySize,
                              (int)smem);
    hmm_kalman_gfx1250<<<1, NT, smem, stream>>>(track, bias_scales,
                                                obs_noise, trans_noise, out);
}